// GCN_71743133712502
// MI455X (gfx1250) — compile-verified
//
#include <hip/hip_runtime.h>

#define HIDDEN 128
#define LDA 132   // padded LDS row stride (floats) -> conflict-free column reads

typedef __attribute__((ext_vector_type(2))) float v2f;
typedef __attribute__((ext_vector_type(8))) float v8f;

// ---------------- degree / dinv ----------------
__global__ void deg_kernel(const int* __restrict__ dst, float* __restrict__ deg, int E) {
    int e = blockIdx.x * blockDim.x + threadIdx.x;
    if (e < E) atomicAdd(&deg[dst[e]], 1.0f);
}

__global__ void dinv_kernel(const float* __restrict__ deg, float* __restrict__ dinv, int N) {
    int i = blockIdx.x * blockDim.x + threadIdx.x;
    if (i < N) dinv[i] = rsqrtf(deg[i] + 1.0f);   // +1 for self loop; always > 0
}

// ---------------- fused LayerNorm + GEMM (xl = LN(x) @ W) ----------------
__global__ __launch_bounds__(256)
void ln_gemm_kernel(const float* __restrict__ x, const float* __restrict__ gamma,
                    const float* __restrict__ beta, const float* __restrict__ W,
                    float* __restrict__ xl, int N) {
    __shared__ float Atile[16 * LDA];
    __shared__ float ssum[16][16];
    __shared__ float ssq[16][16];
    __shared__ float smu[16];
    __shared__ float srs[16];

    const int tid  = threadIdx.x;
    const int lane = tid & 31;
    const int wave = tid >> 5;
    const int r  = tid >> 4;    // tile row this thread helps normalize (0..15)
    const int cg = tid & 15;    // 8-element column group (0..15)

    const int ncol  = wave * 16 + (lane & 15);   // this wave's output column
    const int khalf = (lane >> 4) * 2;           // lanes 16-31 hold K+2,K+3

    // Preload this wave's whole B panel: W[:, ncol] for all 32 k-steps (64 VGPRs)
    v2f breg[32];
#pragma unroll
    for (int kk = 0; kk < 32; ++kk) {
        int k0 = kk * 4 + khalf;
        breg[kk].x = W[(k0    ) * HIDDEN + ncol];
        breg[kk].y = W[(k0 + 1) * HIDDEN + ncol];
    }

    const int row_base = blockIdx.x * 128;
    for (int mt = 0; mt < 8; ++mt) {
        const int row0 = row_base + mt * 16;
        const int grow = row0 + r;

        // --- LayerNorm of 16 rows into LDS A-tile ---
        float vals[8];
        float s = 0.f, sq = 0.f;
#pragma unroll
        for (int j = 0; j < 8; ++j) {
            float v = (grow < N) ? x[(long long)grow * HIDDEN + cg * 8 + j] : 0.f;
            vals[j] = v; s += v; sq += v * v;
        }
        ssum[r][cg] = s; ssq[r][cg] = sq;
        __syncthreads();
        if (cg == 0) {
            float ts = 0.f, tq = 0.f;
#pragma unroll
            for (int j = 0; j < 16; ++j) { ts += ssum[r][j]; tq += ssq[r][j]; }
            float mu  = ts * (1.0f / HIDDEN);
            float var = tq * (1.0f / HIDDEN) - mu * mu;
            smu[r] = mu;
            srs[r] = rsqrtf(var + 1e-5f);
        }
        __syncthreads();
        {
            const float mu = smu[r], rs = srs[r];
#pragma unroll
            for (int j = 0; j < 8; ++j) {
                int c = cg * 8 + j;
                Atile[r * LDA + c] = (vals[j] - mu) * rs * gamma[c] + beta[c];
            }
        }
        __syncthreads();

        // --- D(16x16) = A(16x128) x B(128x16) via 32 f32 WMMAs ---
        v8f acc = {};
        const int arow = lane & 15;
#pragma unroll
        for (int kk = 0; kk < 32; ++kk) {
            int k0 = kk * 4 + khalf;
            v2f a;
            a.x = Atile[arow * LDA + k0];
            a.y = Atile[arow * LDA + k0 + 1];
            acc = __builtin_amdgcn_wmma_f32_16x16x4_f32(false, a, false, breg[kk],
                                                        (short)0, acc, false, false);
        }

        // C/D layout: lanes 0-15 -> M = v, lanes 16-31 -> M = v+8; N = lane&15
        const int mrow0 = row0 + (lane >> 4) * 8;
#pragma unroll
        for (int v = 0; v < 8; ++v) {
            int rr = mrow0 + v;
            if (rr < N) xl[(long long)rr * HIDDEN + ncol] = acc[v];
        }
        __syncthreads();
    }
}

// ---------------- edge scatter: agg[dst] += xl[src] * dinv[src]*dinv[dst] ----------------
__global__ void scatter_kernel(const int* __restrict__ src, const int* __restrict__ dst,
                               const float* __restrict__ dinv, const float* __restrict__ xl,
                               float* __restrict__ agg, int E) {
    long long t = (long long)blockIdx.x * blockDim.x + threadIdx.x;
    int e = (int)(t >> 5);
    if (e >= E) return;
    int c = (int)(t & 31) * 4;
    int s = src[e], d = dst[e];
    float coef = dinv[s] * dinv[d];
    const float4 v = *(const float4*)(xl + (long long)s * HIDDEN + c);
    float* a = agg + (long long)d * HIDDEN + c;
    atomicAdd(a + 0, v.x * coef);
    atomicAdd(a + 1, v.y * coef);
    atomicAdd(a + 2, v.z * coef);
    atomicAdd(a + 3, v.w * coef);
}

// ---------------- finalize: out = relu(agg + xl*dinv^2 + b + x) ----------------
__global__ void finalize_kernel(const float* __restrict__ agg, const float* __restrict__ xl,
                                const float* __restrict__ x, const float* __restrict__ b,
                                const float* __restrict__ dinv, float* __restrict__ out, int N) {
    long long t = (long long)blockIdx.x * blockDim.x + threadIdx.x;
    int i = (int)(t >> 5);
    if (i >= N) return;
    int c = (int)(t & 31) * 4;
    float di = dinv[i];
    float cf = di * di;   // self-loop coefficient
    long long off = (long long)i * HIDDEN + c;
    float4 va = *(const float4*)(agg + off);
    float4 vx = *(const float4*)(xl + off);
    float4 vi = *(const float4*)(x + off);
    float4 vb = *(const float4*)(b + c);
    float4 o;
    o.x = fmaxf(va.x + vx.x * cf + vb.x + vi.x, 0.f);
    o.y = fmaxf(va.y + vx.y * cf + vb.y + vi.y, 0.f);
    o.z = fmaxf(va.z + vx.z * cf + vb.z + vi.z, 0.f);
    o.w = fmaxf(va.w + vx.w * cf + vb.w + vi.w, 0.f);
    *(float4*)(out + off) = o;
}

extern "C" void kernel_launch(void* const* d_in, const int* in_sizes, int n_in,
                              void* d_out, int out_size, void* d_ws, size_t ws_size,
                              hipStream_t stream) {
    // inputs: x, edge_attr, h, gamma, beta, W, b, edge_index, batch
    const float* x     = (const float*)d_in[0];
    const float* h     = (const float*)d_in[2];
    const float* gamma = (const float*)d_in[3];
    const float* beta  = (const float*)d_in[4];
    const float* W     = (const float*)d_in[5];
    const float* b     = (const float*)d_in[6];
    const int*   ei    = (const int*)d_in[7];

    const int N = in_sizes[0] / HIDDEN;
    const int E = in_sizes[7] / 2;
    const int* src = ei;        // edge_index[0]
    const int* dst = ei + E;    // edge_index[1]

    float* ws   = (float*)d_ws;
    float* xl   = ws;                                   // N*H
    float* agg  = xl  + (size_t)N * HIDDEN;             // N*H
    float* deg  = agg + (size_t)N * HIDDEN;             // N
    float* dinv = deg + N;                              // N

    hipMemsetAsync(agg, 0, (size_t)N * HIDDEN * sizeof(float), stream);
    hipMemsetAsync(deg, 0, (size_t)N * sizeof(float), stream);

    deg_kernel <<<(E + 255) / 256, 256, 0, stream>>>(dst, deg, E);
    dinv_kernel<<<(N + 255) / 256, 256, 0, stream>>>(deg, dinv, N);

    ln_gemm_kernel<<<(N + 127) / 128, 256, 0, stream>>>(x, gamma, beta, W, xl, N);

    long long sc_work = (long long)E * 32;
    scatter_kernel<<<(unsigned)((sc_work + 255) / 256), 256, 0, stream>>>(src, dst, dinv, xl, agg, E);

    long long fin_work = (long long)N * 32;
    finalize_kernel<<<(unsigned)((fin_work + 255) / 256), 256, 0, stream>>>(agg, xl, x, b, dinv,
                                                                            (float*)d_out, N);

    // second tuple element: h passthrough
    hipMemcpyAsync((float*)d_out + (size_t)N * HIDDEN, h,
                   (size_t)N * HIDDEN * sizeof(float), hipMemcpyDeviceToDevice, stream);
}